// Multihead_Additive_Attention_24816321036436
// MI455X (gfx1250) — compile-verified
//
#include <hip/hip_runtime.h>
#include <math.h>

typedef __attribute__((ext_vector_type(2))) float v2f;
typedef __attribute__((ext_vector_type(8))) float v8f;

constexpr int B_    = 32;
constexpr int L_    = 2048;
constexpr int D_    = 1024;
constexpr int H_    = 16;
constexpr int DPH_  = 64;    // D / H
constexpr int VPH_  = 128;   // 2D / H
constexpr int TWO_D_ = 2048;

// ---------------------------------------------------------------------------
// Kernel A: Qp = q @ Wq.T   (32 x 1024) = (32 x 1024) x (1024 x 1024)^T
// One wave (32 lanes) per 16x16 output tile, V_WMMA_F32_16X16X4_F32 over K.
// A (16x4 f32): lane(m,hl) VGPR r holds A[m][k + 2*hl + r]  -> float2 load
// B (4x16 f32): lane(m,hl) VGPR r holds B[k + 2*hl + r][m] = Wq[n][k+2*hl+r]
// C (16x16):    VGPR r: lanes0-15 M=r, lanes16-31 M=r+8
// ---------------------------------------------------------------------------
__global__ __launch_bounds__(32) void qproj_wmma(const float* __restrict__ q,
                                                 const float* __restrict__ Wq,
                                                 float* __restrict__ Qp) {
    const int lane = threadIdx.x;
    const int hl   = lane >> 4;      // 0 or 1
    const int m    = lane & 15;
    const int nbase = blockIdx.x * 16;   // 64 tiles over D
    const int mbase = blockIdx.y * 16;   // 2 tiles over B

    const float* qrow = q  + (size_t)(mbase + m) * D_ + 2 * hl;
    const float* wrow = Wq + (size_t)(nbase + m) * D_ + 2 * hl;

    v8f acc = {};
    for (int kk = 0; kk < D_; kk += 4) {
        v2f a = *(const v2f*)(qrow + kk);
        v2f b = *(const v2f*)(wrow + kk);
        acc = __builtin_amdgcn_wmma_f32_16x16x4_f32(false, a, false, b,
                                                    (short)0, acc, false, false);
    }
    #pragma unroll
    for (int r = 0; r < 8; ++r) {
        Qp[(size_t)(mbase + r + 8 * hl) * D_ + nbase + m] = acc[r];
    }
}

// ---------------------------------------------------------------------------
// Kernel B: per (b,h): scores = a1 . tanh(Qp + k), masked softmax over L,
//           ctx = sum_l alpha[l] * v[b,l,head-slice].  Bandwidth-bound part.
// ---------------------------------------------------------------------------
__global__ __launch_bounds__(256) void attn_softmax_ctx(
        const float* __restrict__ Qp, const float* __restrict__ kmat,
        const float* __restrict__ vmat, const int* __restrict__ mask,
        const float* __restrict__ a1, float* __restrict__ alpha,
        float* __restrict__ ctx) {
    __shared__ float qv[DPH_];
    __shared__ float a1s[DPH_];
    __shared__ float sc[L_];           // 8 KB scores / alpha
    __shared__ float red[256];
    __shared__ float ctxp[8 * VPH_];   // 4 KB partial contexts

    const int tid = threadIdx.x;
    const int b = blockIdx.x / H_;
    const int h = blockIdx.x % H_;

    if (tid < DPH_) {
        qv[tid]  = Qp[(size_t)b * D_ + h * DPH_ + tid];
        a1s[tid] = a1[tid];
    }
    __syncthreads();

    // ---- pass 1: scores + running max -------------------------------------
    const float* kb = kmat + (size_t)b * L_ * D_ + (size_t)h * DPH_;
    const int*   mb = mask + (size_t)b * L_;
    float lmax = -3.0e38f;
    for (int i = 0; i < L_ / 256; ++i) {
        const int l = i * 256 + tid;
        const float* kr = kb + (size_t)l * D_;
        float s = 0.f;
        #pragma unroll
        for (int d = 0; d < DPH_; d += 4) {
            float4 k4 = *(const float4*)(kr + d);
            s += a1s[d + 0] * tanhf(qv[d + 0] + k4.x);
            s += a1s[d + 1] * tanhf(qv[d + 1] + k4.y);
            s += a1s[d + 2] * tanhf(qv[d + 2] + k4.z);
            s += a1s[d + 3] * tanhf(qv[d + 3] + k4.w);
        }
        s = (mb[l] != 0) ? s : -1.0e18f;
        sc[l] = s;
        lmax = fmaxf(lmax, s);
    }
    red[tid] = lmax;
    __syncthreads();
    for (int off = 128; off > 0; off >>= 1) {
        if (tid < off) red[tid] = fmaxf(red[tid], red[tid + off]);
        __syncthreads();
    }
    const float gmax = red[0];
    __syncthreads();

    // ---- exp + sum --------------------------------------------------------
    float lsum = 0.f;
    for (int i = 0; i < L_ / 256; ++i) {
        const int l = i * 256 + tid;
        const float e = __expf(sc[l] - gmax);
        sc[l] = e;
        lsum += e;
    }
    red[tid] = lsum;
    __syncthreads();
    for (int off = 128; off > 0; off >>= 1) {
        if (tid < off) red[tid] += red[tid + off];
        __syncthreads();
    }
    const float inv = 1.f / red[0];

    float* arow = alpha + ((size_t)b * H_ + h) * L_;
    for (int i = 0; i < L_ / 256; ++i) {
        const int l = i * 256 + tid;
        const float aup = sc[l] * inv;
        sc[l] = aup;
        arow[l] = aup;
    }
    __syncthreads();

    // ---- pass 2: ctx = sum_l alpha[l] * v ---------------------------------
    // 8 L-partitions x 32 lanes; each lane owns 4 consecutive value comps.
    const int part = tid >> 5;
    const int quad = (tid & 31) * 4;
    const float* vb = vmat + (size_t)b * L_ * TWO_D_ + (size_t)h * VPH_ + quad;
    float ax = 0.f, ay = 0.f, az = 0.f, aw = 0.f;
    const int l0 = part * (L_ / 8);
    for (int j = 0; j < L_ / 8; ++j) {
        const int l = l0 + j;
        __builtin_prefetch((const void*)(vb + (size_t)(l + 16) * TWO_D_), 0, 1);
        const float al = sc[l];
        float4 vv = *(const float4*)(vb + (size_t)l * TWO_D_);
        ax += al * vv.x; ay += al * vv.y; az += al * vv.z; aw += al * vv.w;
    }
    ctxp[part * VPH_ + quad + 0] = ax;
    ctxp[part * VPH_ + quad + 1] = ay;
    ctxp[part * VPH_ + quad + 2] = az;
    ctxp[part * VPH_ + quad + 3] = aw;
    __syncthreads();
    if (tid < VPH_) {
        float s = 0.f;
        #pragma unroll
        for (int p = 0; p < 8; ++p) s += ctxp[p * VPH_ + tid];
        ctx[(size_t)b * TWO_D_ + (size_t)h * VPH_ + tid] = s;
    }
}

// ---------------------------------------------------------------------------
// Kernel C: out = ctx @ Wf.T + bf   (32 x 2048), WMMA f32 16x16x4 again.
// ---------------------------------------------------------------------------
__global__ __launch_bounds__(32) void outproj_wmma(const float* __restrict__ ctx,
                                                   const float* __restrict__ Wf,
                                                   const float* __restrict__ bf,
                                                   float* __restrict__ out) {
    const int lane = threadIdx.x;
    const int hl   = lane >> 4;
    const int m    = lane & 15;
    const int nbase = blockIdx.x * 16;   // 128 tiles over 2D
    const int mbase = blockIdx.y * 16;   // 2 tiles over B

    const float* crow = ctx + (size_t)(mbase + m) * TWO_D_ + 2 * hl;
    const float* wrow = Wf  + (size_t)(nbase + m) * TWO_D_ + 2 * hl;

    v8f acc = {};
    for (int kk = 0; kk < TWO_D_; kk += 4) {
        v2f a = *(const v2f*)(crow + kk);
        v2f b = *(const v2f*)(wrow + kk);
        acc = __builtin_amdgcn_wmma_f32_16x16x4_f32(false, a, false, b,
                                                    (short)0, acc, false, false);
    }
    const float bias = bf[nbase + m];
    #pragma unroll
    for (int r = 0; r < 8; ++r) {
        out[(size_t)(mbase + r + 8 * hl) * TWO_D_ + nbase + m] = acc[r] + bias;
    }
}

// ---------------------------------------------------------------------------
extern "C" void kernel_launch(void* const* d_in, const int* in_sizes, int n_in,
                              void* d_out, int out_size, void* d_ws, size_t ws_size,
                              hipStream_t stream) {
    const float* q    = (const float*)d_in[0];
    const float* v    = (const float*)d_in[1];
    const float* k    = (const float*)d_in[2];
    const int*   msk  = (const int*)  d_in[3];
    const float* Wq   = (const float*)d_in[4];
    const float* a1   = (const float*)d_in[5];
    const float* Wf   = (const float*)d_in[6];
    const float* bf   = (const float*)d_in[7];

    float* alpha = (float*)d_out;                       // [B, H, L]
    float* outv  = alpha + (size_t)B_ * H_ * L_;        // [B, 2D]

    float* Qp  = (float*)d_ws;                          // [B, D]
    float* ctx = Qp + (size_t)B_ * D_;                  // [B, 2D]

    qproj_wmma<<<dim3(D_ / 16, B_ / 16), 32, 0, stream>>>(q, Wq, Qp);
    attn_softmax_ctx<<<dim3(B_ * H_), 256, 0, stream>>>(Qp, k, v, msk, a1,
                                                        alpha, ctx);
    outproj_wmma<<<dim3(TWO_D_ / 16, B_ / 16), 32, 0, stream>>>(ctx, Wf, bf,
                                                                outv);
}